// EmbeddingNet_68118181314966
// MI455X (gfx1250) — compile-verified
//
#include <hip/hip_runtime.h>
#include <hip/hip_bf16.h>

// Problem dims (static in the reference)
#define BATCH 256
#define SEQ   2000
#define EMB   128
#define HALFD 64
#define NPOS  (BATCH * SEQ)   // 512000 flattened (b,s) positions

typedef __attribute__((ext_vector_type(2))) float v2f;
typedef __attribute__((ext_vector_type(8))) float v8f;

// ---------------------------------------------------------------------------
// Kernel 1: per-batch serial pointer chase through LDS.
//   one block per batch row; stage solutions row in LDS, thread 0 chases,
//   everyone writes visited_time (as float) and t = vt % valid to workspace.
// ---------------------------------------------------------------------------
__global__ __launch_bounds__(256) void vt_chase_kernel(
    const int* __restrict__ sol,      // [BATCH, SEQ]
    const int* __restrict__ dy_size,  // scalar
    const int* __restrict__ dy_t,     // scalar
    float* __restrict__ vt_out,       // [BATCH, SEQ] (float-converted int)
    float* __restrict__ t_ws)         // [BATCH, SEQ] t = vt % valid
{
    __shared__ int s_sol[SEQ];
    __shared__ int s_vt[SEQ];

    const int b = blockIdx.x;
    const int* row = sol + (size_t)b * SEQ;

    for (int i = threadIdx.x; i < SEQ; i += blockDim.x) {
        s_sol[i] = row[i];
        s_vt[i]  = 0;
    }
    __syncthreads();

    const int valid = SEQ - dy_size[0] + 2 * dy_t[0];

    if (threadIdx.x == 0) {
        int pre = 0;
        for (int i = 0; i < valid; ++i) {
            int cur = s_sol[pre];
            s_vt[cur] = i + 1;
            pre = cur;
        }
    }
    __syncthreads();

    for (int i = threadIdx.x; i < SEQ; i += blockDim.x) {
        int v = s_vt[i];
        vt_out[(size_t)b * SEQ + i] = (float)v;
        t_ws [(size_t)b * SEQ + i] = (float)(v % valid);
    }
}

// ---------------------------------------------------------------------------
// Kernel 2: fused embedding (WMMA f32 16x16x4, K padded 2->4) + freqs_cis.
//   One wave per 16 flattened positions:
//     A tile : lanes 0-15 hold x[pos][0..1] in VGPR0/1 (K=0,1);
//              lanes 16-31 (K=2,3) are zero padding.
//     B tile : lanes 0-15 hold W[e0+lane][0..1] (rows K=0,1); K=2,3 zero.
//     8 WMMAs sweep EMB=128 in 16-wide blocks, reusing A.
//   Then each lane emits 2 sincos per position for freqs_cis (coalesced
//   float2 stores: 32 lanes x 8B = 256B segments).
// ---------------------------------------------------------------------------
__global__ __launch_bounds__(256) void emb_freqs_kernel(
    const float* __restrict__ x,        // [NPOS, 2]
    const float* __restrict__ W,        // [EMB, 2]
    const float* __restrict__ t_ws,     // [NPOS]
    float* __restrict__ emb_out,        // [NPOS, EMB]
    float* __restrict__ freqs_out)      // [NPOS, HALFD] complex -> float2
{
    const int lane    = threadIdx.x & 31;
    const int wave    = blockIdx.x * (blockDim.x >> 5) + (threadIdx.x >> 5);
    const int posBase = wave * 16;

    // ---- A operand: 16x4 f32 (K=2 real, K=2..3 zero) ----
    v2f a = {0.0f, 0.0f};
    if (lane < 16) {
        const float* xp = x + (size_t)(posBase + lane) * 2;
        a.x = xp[0];        // K=0
        a.y = xp[1];        // K=1
    }

    // ---- 8 WMMA tiles across EMB ----
    #pragma unroll
    for (int eb = 0; eb < 8; ++eb) {
        v2f bmat = {0.0f, 0.0f};
        if (lane < 16) {
            const float* wp = W + (size_t)(eb * 16 + lane) * 2;
            bmat.x = wp[0]; // row K=0, col N=lane
            bmat.y = wp[1]; // row K=1, col N=lane
        }
        v8f c = {};
        c = __builtin_amdgcn_wmma_f32_16x16x4_f32(
                /*neg_a=*/false, a, /*neg_b=*/false, bmat,
                /*c_mod=*/(short)0, c, /*reuse_a=*/false, /*reuse_b=*/false);

        // C layout: lane<16 -> rows 0..7, lane>=16 -> rows 8..15; col = lane&15
        const int n     = lane & 15;
        const int mbase = (lane >> 4) << 3;
        #pragma unroll
        for (int r = 0; r < 8; ++r) {
            emb_out[(size_t)(posBase + mbase + r) * EMB + eb * 16 + n] = c[r];
        }
    }

    // ---- freqs_cis: exp(i * t * theta^(-j/64)), j = 0..63 ----
    const float lnTheta = 9.2103403719761836f;  // ln(10000)
    const float j0 = (float)lane;
    const float j1 = (float)(lane + 32);
    const float f0 = __expf(-(j0 * (1.0f / 64.0f)) * lnTheta);
    const float f1 = __expf(-(j1 * (1.0f / 64.0f)) * lnTheta);

    #pragma unroll 4
    for (int p = 0; p < 16; ++p) {
        const float t = t_ws[posBase + p];      // uniform across wave
        float s0, c0, s1, c1;
        __sincosf(t * f0, &s0, &c0);
        __sincosf(t * f1, &s1, &c1);
        float2* outp = (float2*)(freqs_out + (size_t)(posBase + p) * HALFD * 2);
        outp[lane]      = make_float2(c0, s0);  // (real, imag)
        outp[lane + 32] = make_float2(c1, s1);
    }
}

// ---------------------------------------------------------------------------
// Host launcher
// ---------------------------------------------------------------------------
extern "C" void kernel_launch(void* const* d_in, const int* in_sizes, int n_in,
                              void* d_out, int out_size, void* d_ws, size_t ws_size,
                              hipStream_t stream) {
    const float* x       = (const float*)d_in[0]; // [B,S,2]
    const float* W       = (const float*)d_in[1]; // [EMB,2]
    const int*   sol     = (const int*)  d_in[2]; // [B,S]
    const int*   dy_size = (const int*)  d_in[3]; // scalar
    const int*   dy_t    = (const int*)  d_in[4]; // scalar

    float* out       = (float*)d_out;
    float* emb_out   = out;                                      // NPOS*EMB floats
    float* freqs_out = out + (size_t)NPOS * EMB;                 // NPOS*HALFD*2 floats
    float* vt_out    = out + (size_t)NPOS * EMB + (size_t)NPOS * HALFD * 2;

    float* t_ws = (float*)d_ws;                                  // NPOS floats

    // Kernel 1: pointer chase (one block per batch)
    vt_chase_kernel<<<BATCH, 256, 0, stream>>>(sol, dy_size, dy_t, vt_out, t_ws);

    // Kernel 2: 512000 positions / 16 per wave / 8 waves per block = 4000 blocks
    const int tiles  = NPOS / 16;
    const int blocks = tiles / 8;
    emb_freqs_kernel<<<blocks, 256, 0, stream>>>(x, W, t_ws, emb_out, freqs_out);
}